// TorchAudioEqualizer_40192303956751
// MI455X (gfx1250) — compile-verified
//
#include <hip/hip_runtime.h>

// ---------------- problem constants ----------------
#define TCH    441000         // samples per channel
#define NCH    32             // 16 batch * 2 channels
#define CHUNKS 216            // 2048-sample chunks per channel (216*2048 >= TCH)
#define NTILE  1728           // 16-block (256-sample) tiles per channel = CHUNKS*8
#define TG     16             // tiles per tile-group
#define NTG    108            // tile-groups per channel = NTILE/TG

typedef __attribute__((ext_vector_type(2))) float v2f;
typedef __attribute__((ext_vector_type(8))) float v8f;
typedef __attribute__((ext_vector_type(4))) int   v4i;

// gfx1250 async global->LDS copy path (ASYNCcnt-tracked), with safe fallback.
#if defined(__has_builtin)
#if __has_builtin(__builtin_amdgcn_global_load_async_to_lds_b128) && \
    __has_builtin(__builtin_amdgcn_s_wait_asynccnt)
#define USE_ASYNC_LDS 1
#endif
#endif

#define AS1 __attribute__((address_space(1)))
#define AS3 __attribute__((address_space(3)))

// Peaking-EQ normalized coefficients (torchaudio equalizer_biquad).
__device__ __forceinline__ void eq_coeffs(const float* cfp, const float* gp, const float* qp,
                                          float& b0n, float& b1n, float& b2n,
                                          float& a1n, float& a2n) {
  float cf = fminf(fmaxf(cfp[0], 20.0f), 20000.0f);   // min(20000, 44100/2.1)=20000
  float g  = fminf(fmaxf(gp[0], -30.0f), 30.0f);
  float Q  = fminf(fmaxf(qp[0], 0.1f), 30.0f);
  float w0 = 6.283185307179586f * cf / 44100.0f;
  float sw = __sinf(w0), cw = __cosf(w0);
  float alpha = sw / (2.0f * Q);
  float A = __expf(g * 0.025f * 2.302585092994046f); // 10^(g/40)
  float b0 = 1.0f + alpha * A;
  float b1 = -2.0f * cw;
  float b2 = 1.0f - alpha * A;
  float a0 = 1.0f + alpha / A;
  float a2 = 1.0f - alpha / A;
  float r  = 1.0f / a0;
  b0n = b0 * r; b1n = b1 * r; b2n = b2 * r; a1n = b1 * r; a2n = a2 * r;
}

// h[0..16] of the all-pole impulse response, fully unrolled (stays in regs).
__device__ __forceinline__ void impulse17(float a1n, float a2n, float hr[17]) {
  hr[0] = 1.0f;
  float prev = 0.0f;
  #pragma unroll
  for (int i = 1; i <= 16; ++i) {
    float nh = -a1n * hr[i - 1] - a2n * prev;
    prev = hr[i - 1];
    hr[i] = nh;
  }
}

// Per-block 2x2 state map  s_out = A16 * s_in + (Yz15, Yz14):
// A16 = [[h16, -a2*h15], [h15, -a2*h14]]
__device__ __forceinline__ void a16_mat(float a1n, float a2n,
                                        float& A00, float& A01, float& A10, float& A11) {
  float hr[17];
  impulse17(a1n, a2n, hr);
  A00 = hr[16]; A01 = -a2n * hr[15];
  A10 = hr[15]; A11 = -a2n * hr[14];
}

// ---------------------------------------------------------------------------
// Log-depth inclusive affine scan across the 16 block columns (both lane
// halves mirror the same data).  On entry (u1,u2) = this column's d (with any
// initial state pre-folded into column 0).  On exit (u1,u2) = outgoing state
// of this lane's column:  u_j = A*u_{j-1} + d_j.
// 4 steps x 2 shuffles, combine rule u_new = A^(2^k) * u_from + u_self.
// ---------------------------------------------------------------------------
__device__ __forceinline__ void affine_scan16(int l, int col,
                                              float A00, float A01, float A10, float A11,
                                              float& u1, float& u2) {
  float P00 = A00, P01 = A01, P10 = A10, P11 = A11;
  #pragma unroll
  for (int k = 0; k < 4; ++k) {
    float f1 = __shfl(u1, l - (1 << k), 32);
    float f2 = __shfl(u2, l - (1 << k), 32);
    float n1 = P00 * f1 + P01 * f2 + u1;
    float n2 = P10 * f1 + P11 * f2 + u2;
    bool v = (col >= (1 << k));
    u1 = v ? n1 : u1;
    u2 = v ? n2 : u2;
    if (k < 3) {                               // P <- P^2
      float q00 = P00 * P00 + P01 * P10, q01 = P00 * P01 + P01 * P11;
      float q10 = P10 * P00 + P11 * P10, q11 = P10 * P01 + P11 * P11;
      P00 = q00; P01 = q01; P10 = q10; P11 = q11;
    }
  }
}

// ---------------------------------------------------------------------------
// Shared front end for the two WMMA kernels:
// stage x into LDS (async b128), FIR into stride-17 LDS, Toeplitz T in LDS,
// then per-wave 16x16 tile GEMM  Yz = T @ XF  as 4 chained
// v_wmma_f32_16x16x4_f32.  Returns the D fragment (8 VGPRs).
// D layout: lane l holds rows hi*8..hi*8+7 of block column (l&15).
// ---------------------------------------------------------------------------
__device__ __forceinline__ v8f tile_gemm(
    const float* __restrict__ x, long long chbase, int n0, int tid,
    float b0n, float b1n, float b2n, float a1n, float a2n,
    float* xs, float* xfl, float* hsh, float* Tsh) {
  // ---- phase 1: stage 2048 raw samples (+2 halo); lane 0 writes h[] ----
  {
    int nl = tid * 8;
    int n  = n0 + nl;
    if (n + 7 < TCH) {
#if defined(USE_ASYNC_LDS)
      float* gp0 = (float*)(x + chbase + n);   // builtin wants non-const
      __builtin_amdgcn_global_load_async_to_lds_b128(
          (AS1 v4i*)gp0, (AS3 v4i*)(xs + 2 + nl), 0, 0);
      __builtin_amdgcn_global_load_async_to_lds_b128(
          (AS1 v4i*)(gp0 + 4), (AS3 v4i*)(xs + 2 + nl + 4), 0, 0);
#else
      const float4* p = (const float4*)(x + chbase + n);
      *(float4*)(xs + 2 + nl)     = p[0];
      *(float4*)(xs + 2 + nl + 4) = p[1];
#endif
    } else {
      #pragma unroll
      for (int j = 0; j < 8; ++j) {
        int nn = n + j;
        xs[2 + nl + j] = (nn < TCH) ? x[chbase + nn] : 0.0f;
      }
    }
    if (tid < 2) {
      int nn = n0 - 2 + tid;
      xs[tid] = (nn >= 0 && nn < TCH) ? x[chbase + nn] : 0.0f;
    }
    if (tid == 0) {                   // h[0]=1; h[i] = -a1 h[i-1] - a2 h[i-2]
      float hv = 1.0f, hm = 0.0f;
      #pragma unroll
      for (int i = 0; i < 16; ++i) {
        hsh[i] = hv;
        float hn = -a1n * hv - a2n * hm;
        hm = hv; hv = hn;
      }
    }
  }
#if defined(USE_ASYNC_LDS)
  __builtin_amdgcn_s_wait_asynccnt(0);
#endif
  __syncthreads();

  // ---- phase 2: FIR into xfl (stride 17) + materialize T in LDS ----
  {
    int nl = tid * 8;
    #pragma unroll
    for (int j = 0; j < 8; ++j) {
      int s = nl + j;
      float v = b0n * xs[2 + s] + b1n * xs[1 + s] + b2n * xs[s];
      xfl[(s >> 4) * 17 + (s & 15)] = v;
    }
    int s = tid >> 4, k = tid & 15;   // one T entry per thread
    Tsh[s * 17 + k] = (s >= k) ? hsh[s - k] : 0.0f;
  }
  __syncthreads();

  const int w  = tid >> 5;
  const int l  = tid & 31;
  const int M  = l & 15;
  const int hi = l >> 4;

  // ---- 4 chained WMMAs over K=16; fragments via conflict-free ds_load_b64 ----
  v8f c = {0, 0, 0, 0, 0, 0, 0, 0};
  const float* xfw = xfl + (w * 16) * 17;
  const float* trw = Tsh + M * 17;
  #pragma unroll
  for (int j = 0; j < 4; ++j) {
    const float* ap = trw + 4 * j + 2 * hi;
    v2f a = {ap[0], ap[1]};
    const float* bp = xfw + M * 17 + 4 * j + 2 * hi;
    v2f b = {bp[0], bp[1]};
    c = __builtin_amdgcn_wmma_f32_16x16x4_f32(
            false, a, false, b, (short)0, c, false, false);
  }
  return c;
}

// ---------------------------------------------------------------------------
// K1: tile GEMM, then log-depth compose of the tile's 16 block maps
// (d_b = (row15,row14) of each column live in lanes 16-31 as c[7]/c[6]);
// emit one d per tile (2 floats / 256 samples).
// ---------------------------------------------------------------------------
__global__ __launch_bounds__(256) void eq_tile_d(
    const float* __restrict__ x, const float* __restrict__ cfp,
    const float* __restrict__ gp, const float* __restrict__ qp,
    float* __restrict__ dtile) {
  __shared__ float xs[2050];
  __shared__ float xfl[128 * 17];
  __shared__ float hsh[16];
  __shared__ float Tsh[16 * 17];

  const int tid = threadIdx.x;
  const int ch  = blockIdx.y;
  const long long chbase = (long long)ch * TCH;
  const int n0 = blockIdx.x * 2048;

  float b0n, b1n, b2n, a1n, a2n;
  eq_coeffs(cfp, gp, qp, b0n, b1n, b2n, a1n, a2n);

  v8f c = tile_gemm(x, chbase, n0, tid, b0n, b1n, b2n, a1n, a2n, xs, xfl, hsh, Tsh);

  float A00, A01, A10, A11;
  a16_mat(a1n, a2n, A00, A01, A10, A11);

  const int w = tid >> 5, l = tid & 31, col = l & 15;

  // per-column d, broadcast from lane 16+col; inclusive affine scan (S=0)
  float u1 = __shfl(c[7], 16 + col, 32);
  float u2 = __shfl(c[6], 16 + col, 32);
  affine_scan16(l, col, A00, A01, A10, A11, u1, u2);

  float td1 = __shfl(u1, 15, 32);             // composition over all 16 blocks
  float td2 = __shfl(u2, 15, 32);
  if (l == 0) {
    int t = blockIdx.x * 8 + w;               // tile id within channel
    float2 d2v = {td1, td2};
    *(float2*)(dtile + (ch * NTILE + t) * 2) = d2v;
  }
}

// ---------------------------------------------------------------------------
// K2a: compose TG=16 consecutive tiles -> one d per tile-group.
// Tile map: s_out = A16^16 * s_in + dtile.
// ---------------------------------------------------------------------------
__global__ __launch_bounds__(256) void eq_tg_reduce(
    const float* __restrict__ dtile, const float* __restrict__ cfp,
    const float* __restrict__ gp, const float* __restrict__ qp,
    float* __restrict__ dtg) {
  int t = blockIdx.x * 256 + threadIdx.x;
  if (t >= NCH * NTG) return;
  int ch = t / NTG, tg = t % NTG;

  float b0n, b1n, b2n, a1n, a2n;
  eq_coeffs(cfp, gp, qp, b0n, b1n, b2n, a1n, a2n);
  float P00, P01, P10, P11;
  a16_mat(a1n, a2n, P00, P01, P10, P11);
  #pragma unroll
  for (int i = 0; i < 4; ++i) {               // A16^16
    float q00 = P00 * P00 + P01 * P10, q01 = P00 * P01 + P01 * P11;
    float q10 = P10 * P00 + P11 * P10, q11 = P10 * P01 + P11 * P11;
    P00 = q00; P01 = q01; P10 = q10; P11 = q11;
  }

  const float* dp = dtile + (ch * NTILE + tg * TG) * 2;
  float s1 = 0.f, s2 = 0.f;
  #pragma unroll
  for (int j = 0; j < TG; ++j) {
    float d1 = dp[j * 2], d2 = dp[j * 2 + 1];
    float ns1 = d1 + P00 * s1 + P01 * s2;
    float ns2 = d2 + P10 * s1 + P11 * s2;
    s1 = ns1; s2 = ns2;
  }
  dtg[t * 2] = s1; dtg[t * 2 + 1] = s2;
}

// ---------------------------------------------------------------------------
// K2b: per-channel scan over NTG tile-groups (one lane per channel).
// ---------------------------------------------------------------------------
__global__ void eq_tg_scan(const float* __restrict__ dtg,
                           const float* __restrict__ cfp, const float* __restrict__ gp,
                           const float* __restrict__ qp, float* __restrict__ Stg) {
  int ch = threadIdx.x;
  if (ch >= NCH) return;

  float b0n, b1n, b2n, a1n, a2n;
  eq_coeffs(cfp, gp, qp, b0n, b1n, b2n, a1n, a2n);
  float P00, P01, P10, P11;
  a16_mat(a1n, a2n, P00, P01, P10, P11);
  #pragma unroll
  for (int i = 0; i < 8; ++i) {               // A16^256 (one tile-group)
    float q00 = P00 * P00 + P01 * P10, q01 = P00 * P01 + P01 * P11;
    float q10 = P10 * P00 + P11 * P10, q11 = P10 * P01 + P11 * P11;
    P00 = q00; P01 = q01; P10 = q10; P11 = q11;
  }

  float s1 = 0.f, s2 = 0.f;                    // y[-1]=y[-2]=0
  for (int g = 0; g < NTG; ++g) {
    int idx = ch * NTG + g;
    Stg[idx * 2] = s1; Stg[idx * 2 + 1] = s2;
    float ns1 = P00 * s1 + P01 * s2 + dtg[idx * 2];
    float ns2 = P10 * s1 + P11 * s2 + dtg[idx * 2 + 1];
    s1 = ns1; s2 = ns2;
  }
}

// ---------------------------------------------------------------------------
// K2c: expand tile-group states to per-tile incoming states.
// ---------------------------------------------------------------------------
__global__ __launch_bounds__(256) void eq_tile_expand(
    const float* __restrict__ dtile, const float* __restrict__ Stg,
    const float* __restrict__ cfp, const float* __restrict__ gp,
    const float* __restrict__ qp, float* __restrict__ Stile) {
  int t = blockIdx.x * 256 + threadIdx.x;
  if (t >= NCH * NTG) return;
  int ch = t / NTG, tg = t % NTG;

  float b0n, b1n, b2n, a1n, a2n;
  eq_coeffs(cfp, gp, qp, b0n, b1n, b2n, a1n, a2n);
  float P00, P01, P10, P11;
  a16_mat(a1n, a2n, P00, P01, P10, P11);
  #pragma unroll
  for (int i = 0; i < 4; ++i) {               // A16^16
    float q00 = P00 * P00 + P01 * P10, q01 = P00 * P01 + P01 * P11;
    float q10 = P10 * P00 + P11 * P10, q11 = P10 * P01 + P11 * P11;
    P00 = q00; P01 = q01; P10 = q10; P11 = q11;
  }

  const float* dp = dtile + (ch * NTILE + tg * TG) * 2;
  float* sp = (float*)(Stile + (ch * NTILE + tg * TG) * 2);
  float s1 = Stg[t * 2], s2 = Stg[t * 2 + 1];
  #pragma unroll
  for (int j = 0; j < TG; ++j) {
    sp[j * 2] = s1; sp[j * 2 + 1] = s2;
    float ns1 = dp[j * 2]     + P00 * s1 + P01 * s2;
    float ns2 = dp[j * 2 + 1] + P10 * s1 + P11 * s2;
    s1 = ns1; s2 = ns2;
  }
}

// ---------------------------------------------------------------------------
// K3: recompute the tile GEMM, log-depth scan of within-tile states (initial
// state folded into column 0's d), apply correction, clip, store output.
// y[m, blk] = Yz[m, blk] + s1(blk)*h[m+1] - a2*s2(blk)*h[m]
// ---------------------------------------------------------------------------
__global__ __launch_bounds__(256) void eq_final(
    const float* __restrict__ x, const float* __restrict__ Stile,
    const float* __restrict__ cfp, const float* __restrict__ gp,
    const float* __restrict__ qp, float* __restrict__ out) {
  __shared__ float xs[2050];
  __shared__ float xfl[128 * 17];
  __shared__ float hsh[16];
  __shared__ float Tsh[16 * 17];

  const int tid = threadIdx.x;
  const int ch  = blockIdx.y;
  const long long chbase = (long long)ch * TCH;
  const int n0 = blockIdx.x * 2048;

  float b0n, b1n, b2n, a1n, a2n;
  eq_coeffs(cfp, gp, qp, b0n, b1n, b2n, a1n, a2n);

  v8f c = tile_gemm(x, chbase, n0, tid, b0n, b1n, b2n, a1n, a2n, xs, xfl, hsh, Tsh);

  const int w = tid >> 5, l = tid & 31;
  const int col = l & 15, hi = l >> 4;
  const int t = blockIdx.x * 8 + w;           // tile id within channel

  float hr[17];
  impulse17(a1n, a2n, hr);
  float A00 = hr[16], A01 = -a2n * hr[15], A10 = hr[15], A11 = -a2n * hr[14];

  // per-lane h[m], h[m+1] for its 8 rows m = r + 8*hi (static-index selects)
  float hA[8], hB[8];
  #pragma unroll
  for (int r = 0; r < 8; ++r) {
    hA[r] = hi ? hr[r + 8] : hr[r];
    hB[r] = hi ? hr[r + 9] : hr[r + 1];
  }

  // tile incoming state
  float S1 = Stile[(ch * NTILE + t) * 2];
  float S2 = Stile[(ch * NTILE + t) * 2 + 1];

  // per-column d broadcast; fold A*S into column 0, then inclusive scan
  float u1 = __shfl(c[7], 16 + col, 32);
  float u2 = __shfl(c[6], 16 + col, 32);
  {
    float m1 = A00 * S1 + A01 * S2;
    float m2 = A10 * S1 + A11 * S2;
    u1 = (col == 0) ? u1 + m1 : u1;
    u2 = (col == 0) ? u2 + m2 : u2;
  }
  affine_scan16(l, col, A00, A01, A10, A11, u1, u2);

  // incoming state of this lane's column = previous column's outgoing state
  float p1 = __shfl(u1, l - 1, 32);
  float p2 = __shfl(u2, l - 1, 32);
  float cs1 = (col == 0) ? S1 : p1;
  float cs2 = (col == 0) ? S2 : p2;

  // correction + clip
  float y[8];
  #pragma unroll
  for (int r = 0; r < 8; ++r) {
    float v = c[r] + cs1 * hB[r] - a2n * cs2 * hA[r];
    y[r] = fminf(fmaxf(v, -1.0f), 1.0f);
  }

  // store: lane covers rows hi*8..hi*8+7 of block column col
  {
    int base = n0 + (w * 16 + col) * 16 + hi * 8;
    long long gidx = chbase + base;
    if (base + 7 < TCH) {
      float4 u0 = {y[0], y[1], y[2], y[3]};
      float4 u1v = {y[4], y[5], y[6], y[7]};
      *(float4*)(out + gidx)     = u0;
      *(float4*)(out + gidx + 4) = u1v;
    } else {
      #pragma unroll
      for (int r = 0; r < 8; ++r)
        if (base + r < TCH) out[gidx + r] = y[r];
    }
  }
}

// ---------------------------------------------------------------------------
extern "C" void kernel_launch(void* const* d_in, const int* in_sizes, int n_in,
                              void* d_out, int out_size, void* d_ws, size_t ws_size,
                              hipStream_t stream) {
  const float* x  = (const float*)d_in[0];
  const float* cf = (const float*)d_in[1];
  const float* gn = (const float*)d_in[2];
  const float* q  = (const float*)d_in[3];
  float* out = (float*)d_out;

  float* dtile = (float*)d_ws;                    // NCH*NTILE*2 floats (~442 KB)
  float* dtg   = dtile + NCH * NTILE * 2;         // NCH*NTG*2  floats (~27 KB)
  float* Stg   = dtg   + NCH * NTG * 2;           // NCH*NTG*2  floats (~27 KB)
  float* Stile = Stg   + NCH * NTG * 2;           // NCH*NTILE*2 floats (~442 KB)

  eq_tile_d<<<dim3(CHUNKS, NCH), 256, 0, stream>>>(x, cf, gn, q, dtile);
  eq_tg_reduce<<<(NCH * NTG + 255) / 256, 256, 0, stream>>>(dtile, cf, gn, q, dtg);
  eq_tg_scan<<<1, 32, 0, stream>>>(dtg, cf, gn, q, Stg);
  eq_tile_expand<<<(NCH * NTG + 255) / 256, 256, 0, stream>>>(dtile, Stg, cf, gn, q, Stile);
  eq_final<<<dim3(CHUNKS, NCH), 256, 0, stream>>>(x, Stile, cf, gn, q, out);
}